// TorchHysteresis_75376676045290
// MI455X (gfx1250) — compile-verified
//
#include <hip/hip_runtime.h>
#include <math.h>

typedef __attribute__((ext_vector_type(2))) float v2f;
typedef __attribute__((ext_vector_type(8))) float v8f;

#define TMAX 4096
#define KC 8                        // k-chunks of 4 relays per wave per time-tile
#define PTS_PER_WAVE (4 * KC)       // 32 distinct relays per wave
#define PPL (2 * KC)                // relays tracked per lane = 16
#define WAVES_PER_BLOCK 8
#define BLOCK 256

static __device__ __forceinline__ float bcast_first(float x) {
  return __int_as_float(__builtin_amdgcn_readfirstlane(__float_as_int(x)));
}

// Phase 0: d[j] = softplus(raw[j]) for j < n, 0 for padded tail.
__global__ __launch_bounds__(BLOCK) void hyst_dens_kernel(
    const float* __restrict__ raw, float* __restrict__ dens, int n, int npad) {
  int j = blockIdx.x * blockDim.x + threadIdx.x;
  if (j < npad) {
    float v = 0.0f;
    if (j < n) {
      float x = raw[j];
      v = fmaxf(x, 0.0f) + log1pf(expf(-fabsf(x)));  // stable softplus
    }
    dens[j] = v;
  }
}

// Phase 1: sequential hysteresis scan + WMMA dot-product per 16-step tile.
__global__ __launch_bounds__(BLOCK) void hyst_scan_kernel(
    const float* __restrict__ h, const float* __restrict__ mesh,
    const float* __restrict__ dens, float* __restrict__ partial,
    int T, int n) {
  __shared__ float lh[TMAX];
  for (int i = threadIdx.x; i < T; i += BLOCK) lh[i] = h[i];
  __syncthreads();

  const int lane = threadIdx.x & 31;
  const int row  = lane & 15;       // time-row within 16-tile this lane snapshots
  const int half = lane >> 4;       // K-half: 0 -> relays +0,+1 ; 1 -> +2,+3
  const int w    = blockIdx.x * WAVES_PER_BLOCK + (threadIdx.x >> 5);
  const int pbase = w * PTS_PER_WAVE + 2 * half;

  // Per-lane relay thresholds and densities (kept in registers, fully unrolled).
  float al[PPL], be[PPL], dd[PPL];
#pragma unroll
  for (int kc = 0; kc < KC; ++kc) {
#pragma unroll
    for (int i = 0; i < 2; ++i) {
      int p = pbase + 4 * kc + i;
      float a = 3.0e38f, b = -3.0e38f, d = 0.0f;  // sentinels: never triggered
      if (p < n) { b = mesh[2 * p]; a = mesh[2 * p + 1]; d = dens[p]; }
      al[2 * kc + i] = a; be[2 * kc + i] = b; dd[2 * kc + i] = d;
    }
  }

  unsigned sbits = 0u;     // bit p set <=> state of relay p is +1 (init all -1)
  unsigned snapbits = 0u;  // snapshot of sbits at this lane's time-row
  float* pout = partial + (size_t)w * T;

  float hp = 0.0f;  // h_prev, h_prev[0] = 0
  for (int tb = 0; tb < T; tb += 16) {
    v8f acc = {};
    for (int tt = 0; tt < 16; ++tt) {
      float hi = bcast_first(lh[tb + tt]);  // uniform -> scalar branch
      if (hi > hp) {
#pragma unroll
        for (int p = 0; p < PPL; ++p)
          sbits |= (al[p] <= hi) ? (1u << p) : 0u;
      } else if (hi < hp) {
#pragma unroll
        for (int p = 0; p < PPL; ++p)
          sbits &= (be[p] >= hi) ? ~(1u << p) : 0xffffffffu;
      }
      if (tt == row) snapbits = sbits;  // states[t] = post-update state
      hp = hi;
    }
    // D[m][*] += sum_k A[m][k] * B[k][*] ; A = states(+-1), B = densities
#pragma unroll
    for (int kc = 0; kc < KC; ++kc) {
      v2f A, B;
      A.x = ((snapbits >> (2 * kc)) & 1u) ? 1.0f : -1.0f;
      A.y = ((snapbits >> (2 * kc + 1)) & 1u) ? 1.0f : -1.0f;
      B.x = dd[2 * kc];
      B.y = dd[2 * kc + 1];
      acc = __builtin_amdgcn_wmma_f32_16x16x4_f32(
          false, A, false, B, (short)0, acc, false, false);
    }
    // Every column of D is identical; column 0: lane0 holds M=0..7, lane16 M=8..15
    if (lane == 0) {
      *reinterpret_cast<float4*>(pout + tb) =
          make_float4(acc[0], acc[1], acc[2], acc[3]);
      *reinterpret_cast<float4*>(pout + tb + 4) =
          make_float4(acc[4], acc[5], acc[6], acc[7]);
    } else if (lane == 16) {
      *reinterpret_cast<float4*>(pout + tb + 8) =
          make_float4(acc[0], acc[1], acc[2], acc[3]);
      *reinterpret_cast<float4*>(pout + tb + 12) =
          make_float4(acc[4], acc[5], acc[6], acc[7]);
    }
  }
}

// Phase 2: deterministic column-sum of per-wave partials + affine epilogue.
__global__ __launch_bounds__(BLOCK) void hyst_reduce_kernel(
    const float* __restrict__ partial, const float* __restrict__ scale,
    const float* __restrict__ offset, float* __restrict__ out,
    int T, int NW, float inv_n) {
  int t = blockIdx.x * blockDim.x + threadIdx.x;
  if (t < T) {
    float sum = 0.0f;
    for (int wv = 0; wv < NW; ++wv) sum += partial[(size_t)wv * T + t];
    out[t] = scale[0] * (sum * inv_n) + offset[0];
  }
}

extern "C" void kernel_launch(void* const* d_in, const int* in_sizes, int n_in,
                              void* d_out, int out_size, void* d_ws, size_t ws_size,
                              hipStream_t stream) {
  const float* h      = (const float*)d_in[0];
  const float* mesh   = (const float*)d_in[1];
  const float* raw    = (const float*)d_in[2];
  const float* scale  = (const float*)d_in[3];
  const float* offset = (const float*)d_in[4];
  float* out = (float*)d_out;

  const int T = in_sizes[0];       // 4096 (multiple of 16)
  const int n = in_sizes[2];       // 20100

  const int nWaves  = (n + PTS_PER_WAVE - 1) / PTS_PER_WAVE;
  const int nBlocks = (nWaves + WAVES_PER_BLOCK - 1) / WAVES_PER_BLOCK;
  const int NW      = nBlocks * WAVES_PER_BLOCK;
  const int npad    = NW * PTS_PER_WAVE;

  float* partial = (float*)d_ws;                 // NW * T floats (~10.5 MB)
  float* dens    = partial + (size_t)NW * T;     // npad floats

  hyst_dens_kernel<<<(npad + BLOCK - 1) / BLOCK, BLOCK, 0, stream>>>(
      raw, dens, n, npad);
  hyst_scan_kernel<<<nBlocks, BLOCK, 0, stream>>>(
      h, mesh, dens, partial, T, n);
  hyst_reduce_kernel<<<(T + BLOCK - 1) / BLOCK, BLOCK, 0, stream>>>(
      partial, scale, offset, out, T, NW, 1.0f / (float)n);
}